// HausdorffLoss_37795712204957
// MI455X (gfx1250) — compile-verified
//
#include <hip/hip_runtime.h>
#include <hip/hip_bf16.h>
#include <float.h>

typedef float v2f __attribute__((ext_vector_type(2)));
typedef float v8f __attribute__((ext_vector_type(8)));
typedef int   v4i __attribute__((ext_vector_type(4)));

#define NTOT (4u * 128u * 128u * 128u)   // 8,388,608 elements per volume

#define AS_GLOBAL __attribute__((address_space(1)))
#define AS_LOCAL  __attribute__((address_space(3)))

// ---- CDNA5 async global->LDS copy (guarded so compile never regresses) ----
// Probed signature: (v4i AS_GLOBAL* src, v4i AS_LOCAL* dst, int offset, int cpol)
#if __has_builtin(__builtin_amdgcn_global_load_async_to_lds_b128)
#define HAVE_ASYNC_LDS 1
__device__ __forceinline__ void async_copy_b128(const float* gsrc, float* ldst) {
    __builtin_amdgcn_global_load_async_to_lds_b128(
        (AS_GLOBAL v4i*)gsrc, (AS_LOCAL v4i*)ldst, 0, 0);
}
__device__ __forceinline__ void async_join() {
#if __has_builtin(__builtin_amdgcn_s_wait_asynccnt)
    __builtin_amdgcn_s_wait_asynccnt(0);
#else
    asm volatile("s_wait_asynccnt 0x0" ::: "memory");
#endif
}
#else
#define HAVE_ASYNC_LDS 0
#endif

// wave32 XOR butterfly reduce via ds_swizzle (group-of-32: xor<<10 | and=0x1F)
__device__ __forceinline__ float wave_reduce_add(float s) {
    s += __int_as_float(__builtin_amdgcn_ds_swizzle(__float_as_int(s), 0x401F)); // xor 16
    s += __int_as_float(__builtin_amdgcn_ds_swizzle(__float_as_int(s), 0x201F)); // xor 8
    s += __int_as_float(__builtin_amdgcn_ds_swizzle(__float_as_int(s), 0x101F)); // xor 4
    s += __int_as_float(__builtin_amdgcn_ds_swizzle(__float_as_int(s), 0x081F)); // xor 2
    s += __int_as_float(__builtin_amdgcn_ds_swizzle(__float_as_int(s), 0x041F)); // xor 1
    return s;
}

// ---------------------------------------------------------------------------
// 1) binarize: g = (x >= 0.5) ? 1e12 : 0   (vectorized b128 in / b128 out)
// ---------------------------------------------------------------------------
__global__ void hd_binarize(const float4* __restrict__ x, float4* __restrict__ g, unsigned n4) {
    unsigned i = blockIdx.x * blockDim.x + threadIdx.x;
    unsigned stride = gridDim.x * blockDim.x;
    for (; i < n4; i += stride) {
        float4 v = x[i];
        float4 o;
        o.x = (v.x >= 0.5f) ? 1.0e12f : 0.0f;
        o.y = (v.y >= 0.5f) ? 1.0e12f : 0.0f;
        o.z = (v.z >= 0.5f) ? 1.0e12f : 0.0f;
        o.w = (v.w >= 0.5f) ? 1.0e12f : 0.0f;
        g[i] = o;
    }
}

// ---------------------------------------------------------------------------
// 2) min-plus line pass for strided axes (z: S=16384, y: S=128). In-place.
//    blockIdx.x = 32-wide inner tile (contiguous in memory), blockIdx.y = outer.
//    LDS tile[128][32]; compute lanes all read one row -> conflict-free.
// ---------------------------------------------------------------------------
__global__ void hd_edt_pass_inner(float* __restrict__ g, size_t outerStride, size_t S) {
    __shared__ float tile[128][32];
    const int t  = threadIdx.x;
    const int ti = t & 31;
    const int kg = t >> 5;                     // 0..7
    const size_t base = (size_t)blockIdx.y * outerStride + (size_t)blockIdx.x * 32;

#if HAVE_ASYNC_LDS
    {
        const int tj = t & 7;                  // which quad of the 32-wide inner dim
        const int kq = t >> 3;                 // 0..31
#pragma unroll
        for (int q = 0; q < 4; ++q) {
            int k = kq * 4 + q;
            async_copy_b128(&g[base + (size_t)k * S + tj * 4], &tile[k][tj * 4]);
        }
        async_join();
    }
#else
#pragma unroll
    for (int q = 0; q < 16; ++q) {
        int k = kg * 16 + q;
        tile[k][ti] = g[base + (size_t)k * S + ti];
    }
#endif
    __syncthreads();

    float acc[16];
#pragma unroll
    for (int q = 0; q < 16; ++q) acc[q] = FLT_MAX;
    const float fi0 = (float)(kg * 16);

    for (int j = 0; j < 128; ++j) {
        const float val = tile[j][ti];
        const float fj  = (float)j;
#pragma unroll
        for (int q = 0; q < 16; ++q) {
            float d = (fi0 + (float)q) - fj;
            acc[q] = fminf(acc[q], __fmaf_rn(d, d, val));  // exact d^2+val, matches ref
        }
    }

#pragma unroll
    for (int q = 0; q < 16; ++q) {
        int k = kg * 16 + q;
        g[base + (size_t)k * S + ti] = acc[q];
    }
}

// ---------------------------------------------------------------------------
// 3) min-plus pass along the contiguous axis (S=1). One wave per 128-f line.
// ---------------------------------------------------------------------------
__global__ void hd_edt_pass_x(float* __restrict__ g) {
    __shared__ float tile[8][128];
    const int w    = threadIdx.x >> 5;
    const int lane = threadIdx.x & 31;
    const size_t base = ((size_t)blockIdx.x * 8 + w) * 128;

#if HAVE_ASYNC_LDS
    async_copy_b128(&g[base + lane * 4], &tile[w][lane * 4]);
    async_join();
#else
#pragma unroll
    for (int q = 0; q < 4; ++q)
        tile[w][lane + 32 * q] = g[base + lane + 32 * q];
#endif
    __syncthreads();

    float acc[4] = {FLT_MAX, FLT_MAX, FLT_MAX, FLT_MAX};
    for (int j = 0; j < 128; ++j) {
        const float val = tile[w][j];            // broadcast read
        const float fj  = (float)j;
#pragma unroll
        for (int q = 0; q < 4; ++q) {
            float d = (float)(lane + 32 * q) - fj;
            acc[q] = fminf(acc[q], __fmaf_rn(d, d, val));
        }
    }
#pragma unroll
    for (int q = 0; q < 4; ++q)
        g[base + lane + 32 * q] = acc[q];
}

// ---------------------------------------------------------------------------
// 4) weighted-SSD reduction through V_WMMA_F32_16X16X4_F32.
//    b128 loads on all 4 streams; 128 elements/wave/iter -> 2 WMMAs.
//    Scalar (wave-uniform) trip count: no exec-mask churn around WMMA.
//    Sum over all lanes & C comps == 16 * grand total -> swizzle-reduce, *1/16.
// ---------------------------------------------------------------------------
__global__ void hd_reduce_wmma(const float4* __restrict__ p,  const float4* __restrict__ t,
                               const float4* __restrict__ gp, const float4* __restrict__ gt,
                               float* __restrict__ partials, unsigned niter, unsigned totalWaves) {
    const int tid  = threadIdx.x;
    const int w    = tid >> 5;
    const int lane = tid & 31;
    const unsigned waveId = blockIdx.x * 8 + w;

    v2f ones; ones.x = 1.0f; ones.y = 1.0f;
    v8f c = {};

    for (unsigned it = 0; it < niter; ++it) {        // uniform bound -> s_cbranch
        size_t idx = (size_t)(waveId + it * totalWaves) * 32 + lane;  // float4 units
        float4 pv = p[idx], tv = t[idx], gpv = gp[idx], gtv = gt[idx];
        float d0 = pv.x - tv.x, d1 = pv.y - tv.y, d2 = pv.z - tv.z, d3 = pv.w - tv.w;
        float e0 = d0 * d0 * (gpv.x + gtv.x);
        float e1 = d1 * d1 * (gpv.y + gtv.y);
        float e2 = d2 * d2 * (gpv.z + gtv.z);
        float e3 = d3 * d3 * (gpv.w + gtv.w);
        v2f a0; a0.x = e0; a0.y = e1;
        v2f a1; a1.x = e2; a1.y = e3;
        c = __builtin_amdgcn_wmma_f32_16x16x4_f32(false, a0, false, ones, (short)0, c, false, false);
        c = __builtin_amdgcn_wmma_f32_16x16x4_f32(false, a1, false, ones, (short)0, c, false, false);
    }

    float s = c[0] + c[1] + c[2] + c[3] + c[4] + c[5] + c[6] + c[7];
    s = wave_reduce_add(s);

    __shared__ float wsum[8];
    if (lane == 0) wsum[w] = s * 0.0625f;        // /16 exact
    __syncthreads();
    if (tid == 0) {
        float r = 0.0f;
        for (int k = 0; k < 8; ++k) r += wsum[k];
        partials[blockIdx.x] = r;
    }
}

// ---------------------------------------------------------------------------
// 5) finalize: fixed-order sum of partials, apply 1/N and is_average factor.
// ---------------------------------------------------------------------------
__global__ void hd_finalize(const float* __restrict__ partials, int np,
                            const int* __restrict__ is_avg, float* __restrict__ out,
                            float invN, float batch) {
    __shared__ float buf[256];
    const int tid = threadIdx.x;
    float s = 0.0f;
    for (int i = tid; i < np; i += 256) s += partials[i];
    buf[tid] = s;
    __syncthreads();
    for (int h = 128; h >= 1; h >>= 1) {
        if (tid < h) buf[tid] += buf[tid + h];
        __syncthreads();
    }
    if (tid == 0) {
        float loss = buf[0] * invN;
        if (*is_avg == 0) loss *= batch;
        out[0] = loss;
    }
}

// ---------------------------------------------------------------------------
extern "C" void kernel_launch(void* const* d_in, const int* in_sizes, int n_in,
                              void* d_out, int out_size, void* d_ws, size_t ws_size,
                              hipStream_t stream) {
    const float* pred = (const float*)d_in[0];
    const float* targ = (const float*)d_in[1];
    const int*   isav = (const int*)  d_in[2];

    float* gp       = (float*)d_ws;
    float* gt       = gp + NTOT;
    float* partials = gt + NTOT;      // 1024 floats

    const unsigned N = NTOT;

    // binarize masks (b128)
    hd_binarize<<<4096, 256, 0, stream>>>((const float4*)pred, (float4*)gp, N / 4);
    hd_binarize<<<4096, 256, 0, stream>>>((const float4*)targ, (float4*)gt, N / 4);

    // z pass: outer = b (4), outerStride = 128^3, S = 128^2, 512 inner tiles
    hd_edt_pass_inner<<<dim3(512, 4), 256, 0, stream>>>(gp, (size_t)2097152, (size_t)16384);
    hd_edt_pass_inner<<<dim3(512, 4), 256, 0, stream>>>(gt, (size_t)2097152, (size_t)16384);

    // y pass: outer = (b,z) (512), outerStride = 128^2, S = 128, 4 inner tiles
    hd_edt_pass_inner<<<dim3(4, 512), 256, 0, stream>>>(gp, (size_t)16384, (size_t)128);
    hd_edt_pass_inner<<<dim3(4, 512), 256, 0, stream>>>(gt, (size_t)16384, (size_t)128);

    // x pass: 131072 contiguous lines, 8 per block
    hd_edt_pass_x<<<16384, 256, 0, stream>>>(gp);
    hd_edt_pass_x<<<16384, 256, 0, stream>>>(gt);

    // WMMA reduction: 1024 blocks * 8 waves = 8192 waves, 128 elems/wave/iter
    const unsigned totalWaves = 1024 * 8;
    const unsigned niter      = N / (totalWaves * 128);   // = 8, exact
    hd_reduce_wmma<<<1024, 256, 0, stream>>>((const float4*)pred, (const float4*)targ,
                                             (const float4*)gp, (const float4*)gt,
                                             partials, niter, totalWaves);
    hd_finalize<<<1, 256, 0, stream>>>(partials, 1024, isav, (float*)d_out,
                                       1.0f / (float)N, 2.0f);
}